// PointTransformerLayer_18949395710722
// MI455X (gfx1250) — compile-verified
//
#include <hip/hip_runtime.h>
#include <hip/hip_bf16.h>
#include <math.h>

// ---------------------------------------------------------------------------
// PointTransformerLayer fused for MI455X (gfx1250, wave32).
//   b=2, n=768, dim=64, h=64, hid=256, k=16 neighbors.
// All matrix math via V_WMMA_F32_16X16X4_F32 (CDNA5 native f32 WMMA).
// ---------------------------------------------------------------------------

typedef __attribute__((ext_vector_type(2))) float v2f;
typedef __attribute__((ext_vector_type(8))) float v8f;

#define PT_B   2
#define PT_N   768
#define PT_D   64
#define PT_H   64
#define PT_HID 256
#define PT_K   16

__device__ __forceinline__ v8f wmma4(v2f a, v2f b, v8f c) {
  // D = A(16x4,f32) * B(4x16,f32) + C(16x16,f32)
  return __builtin_amdgcn_wmma_f32_16x16x4_f32(
      /*neg_a=*/false, a, /*neg_b=*/false, b,
      /*c_mod=*/(short)0, c, /*reuse_a=*/false, /*reuse_b=*/false);
}

// ---------------------------------------------------------------------------
// Kernel 1: qkv = x @ w_qkv     (1536x64) @ (64x192) -> (1536x192)
// One wave per 16x16 output tile.  grid = (96, 12), block = 32.
// ---------------------------------------------------------------------------
__global__ void pt_qkv_kernel(const float* __restrict__ X,
                              const float* __restrict__ W,
                              float* __restrict__ Y) {
  const int m0   = blockIdx.x * 16;
  const int c0   = blockIdx.y * 16;
  const int lane = threadIdx.x;
  const int ln   = lane & 15;
  const int hi   = lane >> 4;

  v8f acc = {};
#pragma unroll
  for (int kk = 0; kk < PT_D; kk += 4) {
    const int k0 = kk + 2 * hi;
    v2f a = *(const v2f*)(X + (size_t)(m0 + ln) * PT_D + k0);
    v2f b = { W[(size_t)k0 * 192 + c0 + ln],
              W[(size_t)(k0 + 1) * 192 + c0 + ln] };
    acc = wmma4(a, b, acc);
  }
#pragma unroll
  for (int r = 0; r < 8; ++r)
    Y[(size_t)(m0 + r + 8 * hi) * 192 + c0 + ln] = acc[r];
}

// ---------------------------------------------------------------------------
// Kernel 2: 16-NN per query (squared distance; same ordering as L2 norm).
// grid = 6 (3 blocks per batch), block = 256; batch positions staged in LDS.
// ---------------------------------------------------------------------------
__global__ void pt_knn_kernel(const float* __restrict__ pos,
                              int* __restrict__ nn) {
  __shared__ float sp[PT_N * 3];
  const int batch = blockIdx.x / 3;
  const int i0    = (blockIdx.x % 3) * 256;

  for (int t = threadIdx.x; t < PT_N * 3; t += 256)
    sp[t] = pos[(size_t)batch * PT_N * 3 + t];
  __syncthreads();

  const int i = i0 + threadIdx.x;
  const float px = sp[i * 3 + 0], py = sp[i * 3 + 1], pz = sp[i * 3 + 2];

  float bd[PT_K];
  int   bj[PT_K];
#pragma unroll
  for (int t = 0; t < PT_K; ++t) { bd[t] = 3.4e38f; bj[t] = 0; }

  for (int j = 0; j < PT_N; ++j) {
    const float dx = px - sp[j * 3 + 0];
    const float dy = py - sp[j * 3 + 1];
    const float dz = pz - sp[j * 3 + 2];
    const float d2 = dx * dx + dy * dy + dz * dz;
    if (d2 < bd[PT_K - 1]) {
      bd[PT_K - 1] = d2; bj[PT_K - 1] = j;
#pragma unroll
      for (int t = PT_K - 1; t > 0; --t) {
        if (bd[t] < bd[t - 1]) {
          float td = bd[t]; bd[t] = bd[t - 1]; bd[t - 1] = td;
          int   tj = bj[t]; bj[t] = bj[t - 1]; bj[t - 1] = tj;
        }
      }
    }
  }
  const int gi = batch * PT_N + i;
#pragma unroll
  for (int t = 0; t < PT_K; ++t) nn[gi * PT_K + t] = bj[t];
}

// ---------------------------------------------------------------------------
// Kernel 3: fused per-query attention.  One 4-wave block per (b,i).
// The 16 neighbors form the M=16 dimension of every WMMA GEMM.
// ---------------------------------------------------------------------------
__global__ __launch_bounds__(128) void pt_attn_kernel(
    const float* __restrict__ pos,  const float* __restrict__ qkv,
    const int*   __restrict__ nn,
    const float* __restrict__ pw1,
    const float* __restrict__ pg, const float* __restrict__ pb,
    const float* __restrict__ pm, const float* __restrict__ pv,
    const float* __restrict__ pw2, const float* __restrict__ pb2,
    const float* __restrict__ g1, const float* __restrict__ b1,
    const float* __restrict__ m1, const float* __restrict__ v1,
    const float* __restrict__ w1, const float* __restrict__ bb1,
    const float* __restrict__ g2, const float* __restrict__ b2,
    const float* __restrict__ m2, const float* __restrict__ v2,
    const float* __restrict__ w2, const float* __restrict__ bb2,
    float* __restrict__ out) {
  __shared__ int   sj[PT_K];
  __shared__ float rel[PT_K * 4];
  __shared__ float hph[PT_K * PT_H];     // 16x64
  __shared__ float pe [PT_K * PT_D];     // 16x64
  __shared__ float s1 [PT_K * PT_D];     // 16x64
  __shared__ float vg [PT_K * PT_D];     // 16x64
  __shared__ float s3 [PT_K * PT_HID];   // 16x256
  __shared__ float sim[PT_K * PT_D];     // 16x64
  __shared__ float scp[PT_H], shp[PT_H], sc1[PT_D], sh1[PT_D];
  __shared__ float sc2[PT_HID], sh2[PT_HID];

  const int tid  = threadIdx.x;
  const int lane = tid & 31;
  const int wv   = tid >> 5;
  const int ln   = lane & 15;
  const int hi   = lane >> 4;
  const int bi   = blockIdx.x;
  const int b    = bi / PT_N;
  const int i    = bi % PT_N;

  // Pull the (L2-resident) weight matrices toward this WGP early.
  __builtin_prefetch(w1 + tid * 128, 0, 0);  // emits global_prefetch_b8
  __builtin_prefetch(w2 + tid * 128, 0, 0);

  // Fold BatchNorm (eval) into scale/shift; bn2 also absorbs attn_b1.
  if (tid < 64) {
    float s = pg[tid] * rsqrtf(pv[tid] + 1e-5f);
    scp[tid] = s; shp[tid] = pb[tid] - pm[tid] * s;
    float t = g1[tid] * rsqrtf(v1[tid] + 1e-5f);
    sc1[tid] = t; sh1[tid] = b1[tid] - m1[tid] * t;
  }
  for (int c = tid; c < PT_HID; c += 128) {
    float s = g2[c] * rsqrtf(v2[c] + 1e-5f);
    sc2[c] = s; sh2[c] = b2[c] - m2[c] * s + bb1[c] * s;
  }
  if (tid < PT_K) {
    const int j = nn[bi * PT_K + tid];
    sj[tid] = j;
#pragma unroll
    for (int t = 0; t < 3; ++t)
      rel[tid * 4 + t] = pos[(size_t)(b * PT_N + i) * 3 + t] -
                         pos[(size_t)(b * PT_N + j) * 3 + t];
  }
  __syncthreads();

  // hph = relu(bn(rel @ pos_w1))  : 16x64, VALU (K=3 too small for WMMA)
  {
    const int jn = tid >> 3;
    const int cb = (tid & 7) * 8;
    const float r0 = rel[jn * 4 + 0], r1 = rel[jn * 4 + 1], r2 = rel[jn * 4 + 2];
#pragma unroll
    for (int e = 0; e < 8; ++e) {
      const int c = cb + e;
      float d = r0 * pw1[c] + r1 * pw1[PT_H + c] + r2 * pw1[2 * PT_H + c];
      d = d * scp[c] + shp[c];
      hph[jn * PT_H + c] = d > 0.f ? d : 0.f;
    }
  }
  __syncthreads();

  // pe = hph @ pos_w2 + pos_b2 : 16x64, wave wv owns cols [16wv,16wv+16)
  {
    const int c0 = 16 * wv;
    v8f acc = {};
#pragma unroll
    for (int kk = 0; kk < PT_H; kk += 4) {
      const int k0 = kk + 2 * hi;
      v2f a = *(const v2f*)(&hph[ln * PT_H + k0]);
      v2f bv = { pw2[k0 * PT_D + c0 + ln], pw2[(k0 + 1) * PT_D + c0 + ln] };
      acc = wmma4(a, bv, acc);
    }
#pragma unroll
    for (int r = 0; r < 8; ++r) {
      const int row = r + 8 * hi, col = c0 + ln;
      pe[row * PT_D + col] = acc[r] + pb2[col];
    }
  }
  __syncthreads();

  // s1 = relu(bn1((q_i - k_j) + pe)),  vg = v_j + pe
  {
    const int jn = tid >> 3;
    const int j  = sj[jn];
    const int cb = (tid & 7) * 8;
    const float* qrow = qkv + (size_t)(b * PT_N + i) * 192;
    const float* krow = qkv + (size_t)(b * PT_N + j) * 192 + 64;
    const float* vrow = qkv + (size_t)(b * PT_N + j) * 192 + 128;
#pragma unroll
    for (int e = 0; e < 8; ++e) {
      const int c = cb + e;
      const float p  = pe[jn * PT_D + c];
      const float s0 = (qrow[c] - krow[c]) + p;
      const float t  = s0 * sc1[c] + sh1[c];
      s1[jn * PT_D + c] = t > 0.f ? t : 0.f;
      vg[jn * PT_D + c] = vrow[c] + p;
    }
  }
  __syncthreads();

  // s3 = relu(bn2(s1 @ attn_w1 + b1)) : 16x256, wave wv owns cols [64wv,64wv+64)
  for (int t = 0; t < 4; ++t) {
    const int c0 = 64 * wv + 16 * t;
    v8f acc = {};
#pragma unroll
    for (int kk = 0; kk < PT_D; kk += 4) {
      const int k0 = kk + 2 * hi;
      v2f a = *(const v2f*)(&s1[ln * PT_D + k0]);
      v2f bv = { w1[k0 * PT_HID + c0 + ln], w1[(k0 + 1) * PT_HID + c0 + ln] };
      acc = wmma4(a, bv, acc);
    }
#pragma unroll
    for (int r = 0; r < 8; ++r) {
      const int row = r + 8 * hi, col = c0 + ln;
      const float x = acc[r] * sc2[col] + sh2[col];
      s3[row * PT_HID + col] = x > 0.f ? x : 0.f;
    }
  }
  __syncthreads();

  // sim = s3 @ attn_w2 + b2 : 16x64, wave wv owns cols [16wv,16wv+16), K=256
  {
    const int c0 = 16 * wv;
    v8f acc = {};
#pragma unroll 8
    for (int kk = 0; kk < PT_HID; kk += 4) {
      const int k0 = kk + 2 * hi;
      v2f a = *(const v2f*)(&s3[ln * PT_HID + k0]);
      v2f bv = { w2[k0 * PT_D + c0 + ln], w2[(k0 + 1) * PT_D + c0 + ln] };
      acc = wmma4(a, bv, acc);
    }
#pragma unroll
    for (int r = 0; r < 8; ++r) {
      const int row = r + 8 * hi, col = c0 + ln;
      sim[row * PT_D + col] = acc[r] + bb2[col];
    }
  }
  __syncthreads();

  // softmax over the 16 neighbors (per channel) + weighted aggregate of vg
  if (tid < PT_D) {
    const int c = tid;
    float mx = -3.4e38f;
#pragma unroll
    for (int jn = 0; jn < PT_K; ++jn) mx = fmaxf(mx, sim[jn * PT_D + c]);
    float den = 0.f, num = 0.f;
#pragma unroll
    for (int jn = 0; jn < PT_K; ++jn) {
      const float e = __expf(sim[jn * PT_D + c] - mx);
      den += e;
      num += e * vg[jn * PT_D + c];
    }
    out[(size_t)(b * PT_N + i) * PT_D + c] = num / den;
  }
}

// ---------------------------------------------------------------------------
extern "C" void kernel_launch(void* const* d_in, const int* in_sizes, int n_in,
                              void* d_out, int out_size, void* d_ws, size_t ws_size,
                              hipStream_t stream) {
  const float* x     = (const float*)d_in[0];
  const float* pos   = (const float*)d_in[1];
  const float* w_qkv = (const float*)d_in[2];
  const float* pw1   = (const float*)d_in[3];
  const float* pg    = (const float*)d_in[4];
  const float* pb    = (const float*)d_in[5];
  const float* pm    = (const float*)d_in[6];
  const float* pv    = (const float*)d_in[7];
  const float* pw2   = (const float*)d_in[8];
  const float* pb2   = (const float*)d_in[9];
  const float* g1    = (const float*)d_in[10];
  const float* b1    = (const float*)d_in[11];
  const float* m1    = (const float*)d_in[12];
  const float* v1    = (const float*)d_in[13];
  const float* w1    = (const float*)d_in[14];
  const float* bb1   = (const float*)d_in[15];
  const float* g2    = (const float*)d_in[16];
  const float* b2    = (const float*)d_in[17];
  const float* m2    = (const float*)d_in[18];
  const float* v2    = (const float*)d_in[19];
  const float* w2    = (const float*)d_in[20];
  const float* bb2   = (const float*)d_in[21];
  float* out = (float*)d_out;

  const int rows = PT_B * PT_N;                       // 1536
  float* qkv = (float*)d_ws;                          // rows*192 floats
  int*   nn  = (int*)((char*)d_ws + (size_t)rows * 192 * sizeof(float));

  pt_qkv_kernel<<<dim3(rows / 16, 192 / 16), 32, 0, stream>>>(x, w_qkv, qkv);
  pt_knn_kernel<<<PT_B * 3, 256, 0, stream>>>(pos, nn);
  pt_attn_kernel<<<rows, 128, 0, stream>>>(
      pos, qkv, nn, pw1, pg, pb, pm, pv, pw2, pb2,
      g1, b1, m1, v1, w1, bb1, g2, b2, m2, v2, w2, bb2, out);
}